// JKNetMaxpool_34832184770813
// MI455X (gfx1250) — compile-verified
//
#include <hip/hip_runtime.h>
#include <math.h>

typedef __attribute__((ext_vector_type(2))) float v2f;
typedef __attribute__((ext_vector_type(8))) float v8f;

// ---------------------------------------------------------------------------
// Zero-fill (grid-stride)
// ---------------------------------------------------------------------------
__global__ void zero_f32(float* __restrict__ p, size_t n) {
    size_t i = (size_t)blockIdx.x * blockDim.x + threadIdx.x;
    size_t stride = (size_t)gridDim.x * blockDim.x;
    for (; i < n; i += stride) p[i] = 0.0f;
}

// ---------------------------------------------------------------------------
// Degree accumulation: deg_out[src[e]] += 1, deg_in[dst[e]] += 1
// ---------------------------------------------------------------------------
__global__ void degrees_kernel(const int* __restrict__ src, const int* __restrict__ dst,
                               float* __restrict__ degO, float* __restrict__ degI, int nE) {
    int i = blockIdx.x * blockDim.x + threadIdx.x;
    if (i < nE) {
        atomicAdd(&degO[src[i]], 1.0f);
        atomicAdd(&degI[dst[i]], 1.0f);
    }
}

// deg -> (deg > 0 ? rsqrt(deg) : 0), applied in place to 2N contiguous floats
__global__ void make_norm(float* __restrict__ d, int n) {
    int i = blockIdx.x * blockDim.x + threadIdx.x;
    if (i < n) {
        float v = d[i];
        d[i] = (v > 0.0f) ? rsqrtf(v) : 0.0f;
    }
}

// ---------------------------------------------------------------------------
// GEMM with fused row scaling:  C[M x Nout] = (A * out_norm[row]) @ W
// A: M x K row-major, W: K x Nout row-major, C: M x Nout row-major.
// One block per 16-row M tile; each wave computes 16 x (16*NT) of C using
// V_WMMA_F32_16X16X4_F32.  GUARD enables column bounds checks (Nout=40 case).
// blockDim.x = 32 * ceil(Nout / (16*NT))  (Nout multiple of 16*NT when !GUARD)
// ---------------------------------------------------------------------------
template <int NT, bool GUARD>
__global__ void gemm_norm_wmma(const float* __restrict__ A, int K,
                               const float* __restrict__ outn,
                               const float* __restrict__ W, int Nout,
                               float* __restrict__ C) {
    const int lane  = threadIdx.x & 31;
    const int wave  = threadIdx.x >> 5;
    const int m0    = blockIdx.x * 16;
    const int nbase = wave * (16 * NT);
    const int half  = lane & 15;      // N (or M) index within tile
    const bool hi   = lane >= 16;
    const int rowA  = m0 + half;
    const int koff  = hi ? 2 : 0;

    const float nrm = outn[rowA];
    const float* __restrict__ Arow = A + (size_t)rowA * K;

    v8f zero = {};
    v8f acc[NT];
#pragma unroll
    for (int t = 0; t < NT; ++t) acc[t] = zero;

    for (int k0 = 0; k0 < K; k0 += 4) {
        v2f a;
        a[0] = Arow[k0 + koff] * nrm;        // lanes 0-15: K=k0,k0+1 ; 16-31: K=k0+2,k0+3
        a[1] = Arow[k0 + koff + 1] * nrm;
        const float* __restrict__ Wk = W + (size_t)(k0 + koff) * Nout;
#pragma unroll
        for (int t = 0; t < NT; ++t) {
            int col = nbase + t * 16 + half;
            v2f b;
            if (!GUARD || col < Nout) {
                b[0] = Wk[col];
                b[1] = Wk[col + Nout];
            } else {
                b[0] = 0.0f;
                b[1] = 0.0f;
            }
            acc[t] = __builtin_amdgcn_wmma_f32_16x16x4_f32(
                false, a, false, b, (short)0, acc[t], false, false);
        }
    }

#pragma unroll
    for (int t = 0; t < NT; ++t) {
        int col = nbase + t * 16 + half;
        if (GUARD && col >= Nout) continue;
#pragma unroll
        for (int j = 0; j < 8; ++j) {
            int row = m0 + j + (hi ? 8 : 0);   // VGPR j: lanes 0-15 -> M=j, 16-31 -> M=j+8
            C[(size_t)row * Nout + col] = acc[t][j];
        }
    }
}

// ---------------------------------------------------------------------------
// Edge scatter-add: agg[dst[e]][:] += proj[src[e]][:]   (one wave per edge)
// ---------------------------------------------------------------------------
__global__ void scatter_add(const float* __restrict__ proj, const int* __restrict__ src,
                            const int* __restrict__ dst, float* __restrict__ agg,
                            int nE, int F) {
    int e = blockIdx.x * (blockDim.x >> 5) + (threadIdx.x >> 5);
    if (e >= nE) return;
    int lane = threadIdx.x & 31;
    int s = src[e];
    int d = dst[e];
    const float* __restrict__ p = proj + (size_t)s * F;
    float* __restrict__ a       = agg  + (size_t)d * F;
    for (int f = lane; f < F; f += 32)
        atomicAdd(&a[f], p[f]);
}

// ---------------------------------------------------------------------------
// Post pass (F = 256): h = relu(h * in_norm + bias); jk = first ? h : max(jk,h)
// ---------------------------------------------------------------------------
__global__ void post_relu_jk(float* __restrict__ h, const float* __restrict__ inn,
                             const float* __restrict__ bias, float* __restrict__ jk,
                             size_t total, int first) {
    size_t idx = (size_t)blockIdx.x * blockDim.x + threadIdx.x;
    if (idx >= total) return;
    int node = (int)(idx >> 8);     // F == 256
    int f    = (int)(idx & 255);
    float v = fmaxf(h[idx] * inn[node] + bias[f], 0.0f);
    h[idx]  = v;
    jk[idx] = first ? v : fmaxf(jk[idx], v);
}

// ---------------------------------------------------------------------------
// Final: z = agg * in_norm + bias (40-wide, no relu), then log_softmax per node
// ---------------------------------------------------------------------------
__global__ void final_lsm(const float* __restrict__ agg, const float* __restrict__ inn,
                          const float* __restrict__ bias, float* __restrict__ out, int N) {
    int n = blockIdx.x * blockDim.x + threadIdx.x;
    if (n >= N) return;
    const float innv = inn[n];
    float z[40];
    float m = -INFINITY;
#pragma unroll
    for (int f = 0; f < 40; ++f) {
        float v = agg[(size_t)n * 40 + f] * innv + bias[f];
        z[f] = v;
        m = fmaxf(m, v);
    }
    float s = 0.0f;
#pragma unroll
    for (int f = 0; f < 40; ++f) s += expf(z[f] - m);
    float l = logf(s);
#pragma unroll
    for (int f = 0; f < 40; ++f) out[(size_t)n * 40 + f] = z[f] - m - l;
}

// ---------------------------------------------------------------------------
// Host-side orchestration (all on `stream`; graph-capture safe)
// ---------------------------------------------------------------------------
extern "C" void kernel_launch(void* const* d_in, const int* in_sizes, int n_in,
                              void* d_out, int out_size, void* d_ws, size_t ws_size,
                              hipStream_t stream) {
    const float* feats = (const float*)d_in[0];   // [N, 512]
    const int*   src   = (const int*)d_in[1];     // [E]
    const int*   dst   = (const int*)d_in[2];     // [E]
    const float* W0    = (const float*)d_in[3];   // [512, 256]
    const float* b0    = (const float*)d_in[4];   // [256]
    const float* Ws    = (const float*)d_in[5];   // [5, 256, 256]
    const float* bs    = (const float*)d_in[6];   // [5, 256]
    const float* Wout  = (const float*)d_in[7];   // [256, 40]
    const float* bout  = (const float*)d_in[8];   // [40]
    float*       out   = (float*)d_out;           // [N, 40]

    const int N   = in_sizes[0] / 512;            // 100000 (divisible by 16)
    const int E   = in_sizes[1];                  // 1600000
    const int F   = 256;
    const int CLS = 40;

    // Workspace carve-up
    char* ws = (char*)d_ws;
    float* degO = (float*)ws;                     // N floats -> out_norm
    float* degI = degO + N;                       // N floats -> in_norm
    size_t normBytes = ((size_t)2 * N * sizeof(float) + 255) & ~(size_t)255;
    const size_t big = (size_t)N * F;
    float* X  = (float*)(ws + normBytes);         // N x 256
    float* Y  = X + big;                          // N x 256
    float* JK = Y + big;                          // N x 256

    const int mTiles = N / 16;                    // 6250

    // ---- degrees & norms ----
    zero_f32<<<2048, 256, 0, stream>>>(degO, (size_t)2 * N);
    degrees_kernel<<<(E + 255) / 256, 256, 0, stream>>>(src, dst, degO, degI, E);
    make_norm<<<(2 * N + 255) / 256, 256, 0, stream>>>(degO, 2 * N);

    const int postBlocks    = (int)((big + 255) / 256);
    const int scatterBlocks = (E + 7) / 8;        // 8 edges (waves) per 256-thr block

    // ---- layer 0: features(512) -> 256 ----
    gemm_norm_wmma<4, false><<<mTiles, 128, 0, stream>>>(feats, 512, degO, W0, F, Y);
    zero_f32<<<2048, 256, 0, stream>>>(X, big);
    scatter_add<<<scatterBlocks, 256, 0, stream>>>(Y, src, dst, X, E, F);
    post_relu_jk<<<postBlocks, 256, 0, stream>>>(X, degI, b0, JK, big, 1);

    // ---- layers 1..5: 256 -> 256 ----
    for (int i = 0; i < 5; ++i) {
        const float* Wi = Ws + (size_t)i * F * F;
        const float* bi = bs + (size_t)i * F;
        gemm_norm_wmma<4, false><<<mTiles, 128, 0, stream>>>(X, F, degO, Wi, F, Y);
        zero_f32<<<2048, 256, 0, stream>>>(X, big);
        scatter_add<<<scatterBlocks, 256, 0, stream>>>(Y, src, dst, X, E, F);
        post_relu_jk<<<postBlocks, 256, 0, stream>>>(X, degI, bi, JK, big, 0);
    }

    // ---- output layer: JK(256) -> 40, then log_softmax ----
    gemm_norm_wmma<3, true><<<mTiles, 32, 0, stream>>>(JK, F, degO, Wout, CLS, Y);
    zero_f32<<<2048, 256, 0, stream>>>(X, (size_t)N * CLS);
    scatter_add<<<scatterBlocks, 256, 0, stream>>>(Y, src, dst, X, E, CLS);
    final_lsm<<<(N + 255) / 256, 256, 0, stream>>>(X, degI, bout, out, N);
}